// AdjStackAttentionWeights_40458591928609
// MI455X (gfx1250) — compile-verified
//
#include <hip/hip_runtime.h>
#include <hip/hip_bf16.h>
#include <stdint.h>

#define B_ 64
#define N_ 256
#define S_ 16
#define HID_ 32
#define HEADS_ 8
#define BN_EPS 1e-5f

typedef __attribute__((ext_vector_type(2))) float v2f;
typedef __attribute__((ext_vector_type(8))) float v8f;

// workspace layout (floats)
#define WS_SX   0     // [16] sum x
#define WS_SX2  16    // [16] sum x^2
#define WS_W1F  32    // [16][32] BN-folded W1
#define WS_B1F  544   // [32]     BN-folded b1
#define WS_W2P  576   // [32][16] W2 padded to N=16
#define WS_B2P  1088  // [16]     b2 padded

// ---------------- pass 1: masked BN statistics (b128 loads) ----------------
__global__ __launch_bounds__(256) void stats_kernel(
    const float* __restrict__ stacks, const uint8_t* __restrict__ mask,
    float* __restrict__ ws) {
  __shared__ float red[2][S_];
  int tid = threadIdx.x;
  int q = tid & 3;                           // channel quad: s = 4q..4q+3
  long group   = ((long)blockIdx.x * blockDim.x + tid) >> 2;
  long gstride = ((long)gridDim.x * blockDim.x) >> 2;
  const long P = (long)B_ * N_ * N_;
  float acc[4] = {0.f, 0.f, 0.f, 0.f}, acc2[4] = {0.f, 0.f, 0.f, 0.f};
  for (long p = group; p < P; p += gstride) {
    int bn = (int)(p >> 8);                  // b*N + n (m = low 8 bits)
    if (!mask[bn]) {                         // valid row
      float4 x = *reinterpret_cast<const float4*>(stacks + (p << 4) + q * 4);
      acc[0] += x.x; acc2[0] += x.x * x.x;
      acc[1] += x.y; acc2[1] += x.y * x.y;
      acc[2] += x.z; acc2[2] += x.z * x.z;
      acc[3] += x.w; acc2[3] += x.w * x.w;
    }
  }
  if (tid < 32) red[tid >> 4][tid & 15] = 0.f;
  __syncthreads();
#pragma unroll
  for (int i = 0; i < 4; i++) {
    atomicAdd(&red[0][q * 4 + i], acc[i]);
    atomicAdd(&red[1][q * 4 + i], acc2[i]);
  }
  __syncthreads();
  if (tid < 16) {
    atomicAdd(&ws[WS_SX  + tid], red[0][tid]);
    atomicAdd(&ws[WS_SX2 + tid], red[1][tid]);
  }
}

// ---------------- pass 2: fold BN into layer-1 weights ----------------
__global__ __launch_bounds__(256) void finalize_kernel(
    const uint8_t* __restrict__ mask, const float* __restrict__ gamma,
    const float* __restrict__ beta, const float* __restrict__ W1,
    const float* __restrict__ b1, const float* __restrict__ W2,
    const float* __restrict__ b2, float* __restrict__ ws) {
  __shared__ float cred[256];
  __shared__ float scale[16], shift[16];
  int tid = threadIdx.x;
  float c = 0.f;
  for (int i = tid; i < B_ * N_; i += 256) c += mask[i] ? 0.f : 1.f;
  cred[tid] = c;
  __syncthreads();
  for (int off = 128; off > 0; off >>= 1) {
    if (tid < off) cred[tid] += cred[tid + off];
    __syncthreads();
  }
  float count = cred[0] * (float)N_;         // K * n valid BN rows
  if (tid < 16) {
    float m  = ws[WS_SX + tid] / count;
    float v  = ws[WS_SX2 + tid] / count - m * m;
    float sc = gamma[tid] * rsqrtf(v + BN_EPS);
    scale[tid] = sc;
    shift[tid] = beta[tid] - m * sc;
  }
  __syncthreads();
  if (tid < HID_) {
    float acc = b1[tid];
    for (int s = 0; s < S_; s++) {
      float w = W1[s * HID_ + tid];
      ws[WS_W1F + s * HID_ + tid] = scale[s] * w;
      acc += shift[s] * w;
    }
    ws[WS_B1F + tid] = acc;
  }
  for (int i = tid; i < HID_ * 16; i += 256) {
    int k = i >> 4, n = i & 15;
    ws[WS_W2P + i] = (n < HEADS_) ? W2[k * HEADS_ + n] : 0.f;
  }
  if (tid < 16) ws[WS_B2P + tid] = (tid < HEADS_) ? b2[tid] : 0.f;
}

// ---------------- pass 3: row-persistent fused MLP via f32 WMMA ----------------
// One wave owns one (b,n) row: loads all WMMA B-fragments once, then streams
// 16 A-tiles (1 KB contiguous each) through 16 chained 16x16x4 f32 WMMAs per
// tile. Masked rows skip all compute and zero-fill the output. Results are
// packed through LDS into one global_store_b128 per lane per tile.
__global__ __launch_bounds__(256) void fused_mlp_kernel(
    const float* __restrict__ stacks, const uint8_t* __restrict__ mask,
    const float* __restrict__ ws, float* __restrict__ out) {
  __shared__ float lds[8][672];              // per wave: h[16*34] + y[128]
  int lane = threadIdx.x & 31;
  int wv   = threadIdx.x >> 5;
  int hi   = lane >> 4;                      // upper half-wave -> K+2 / M+8
  int lm   = lane & 15;
  int bn   = blockIdx.x * 8 + wv;            // 16384 rows total
  float* ob = out + (long)bn * (N_ * HEADS_);

  if (mask[bn]) {                            // padded row -> pure zero-fill
    float4 z = {0.f, 0.f, 0.f, 0.f};
#pragma unroll
    for (int t = 0; t < 16; t++)
      *reinterpret_cast<float4*>(ob + t * 128 + lane * 4) = z;
    return;
  }

  // ---- load all weight fragments / biases once per wave ----
  v2f bw1[2][4];                             // GEMM1 B frags [N-half][K-slice]
  float bi1[2];
#pragma unroll
  for (int j = 0; j < 2; j++) {
    bi1[j] = ws[WS_B1F + 16 * j + lm];
#pragma unroll
    for (int q = 0; q < 4; q++) {
      int k = 4 * q + 2 * hi;
      bw1[j][q][0] = ws[WS_W1F + k * HID_ + 16 * j + lm];
      bw1[j][q][1] = ws[WS_W1F + (k + 1) * HID_ + 16 * j + lm];
    }
  }
  v2f bw2[8];                                // GEMM2 B frags [K-slice]
  float bi2 = ws[WS_B2P + lm];
#pragma unroll
  for (int q = 0; q < 8; q++) {
    int k = 4 * q + 2 * hi;
    bw2[q][0] = ws[WS_W2P + k * 16 + lm];
    bw2[q][1] = ws[WS_W2P + (k + 1) * 16 + lm];
  }

  const float* xb = stacks + (long)bn * (N_ * S_);
  float* L  = &lds[wv][0];
  float* yb = L + 544;

  for (int mt = 0; mt < 16; mt++) {
    const float* xt = xb + mt * 256;
    if (mt < 15)                             // prefetch next 1 KB A tile
      __builtin_prefetch(xt + 256 + lane * 8, 0, 0);

    // A fragments: 4 K-slices of f32 16x4 (V0=K even, V1=K odd; hi -> K+2)
    v2f a[4];
#pragma unroll
    for (int q = 0; q < 4; q++) {
      float2 t = *reinterpret_cast<const float2*>(xt + lm * S_ + 4 * q + 2 * hi);
      a[q][0] = t.x; a[q][1] = t.y;
    }

    // GEMM1: h = relu(x @ W1' + b1')   two independent WMMA chains
    v8f c0, c1;
#pragma unroll
    for (int i = 0; i < 8; i++) { c0[i] = bi1[0]; c1[i] = bi1[1]; }
#pragma unroll
    for (int q = 0; q < 4; q++) {
      c0 = __builtin_amdgcn_wmma_f32_16x16x4_f32(
          false, a[q], false, bw1[0][q], (short)0, c0, false, false);
      c1 = __builtin_amdgcn_wmma_f32_16x16x4_f32(
          false, a[q], false, bw1[1][q], (short)0, c1, false, false);
    }
#pragma unroll
    for (int i = 0; i < 8; i++) {
      c0[i] = fmaxf(c0[i], 0.f);
      c1[i] = fmaxf(c1[i], 0.f);
    }

    // stage h[16][32] to LDS (C/D layout -> row-major, stride 34: no conflicts)
#pragma unroll
    for (int v = 0; v < 8; v++) {
      L[(v + 8 * hi) * 34 + lm]      = c0[v];
      L[(v + 8 * hi) * 34 + 16 + lm] = c1[v];
    }
    asm volatile("s_wait_dscnt 0" ::: "memory");   // wave-synchronous LDS RAW

    // GEMM2: y = h @ W2p + b2p   (K=32 -> 8 chained WMMAs)
    v8f c2;
#pragma unroll
    for (int i = 0; i < 8; i++) c2[i] = bi2;
#pragma unroll
    for (int q = 0; q < 8; q++) {
      int k = 4 * q + 2 * hi;
      v2f a2;
      float2 t = *reinterpret_cast<const float2*>(L + lm * 34 + k);
      a2[0] = t.x; a2[1] = t.y;
      c2 = __builtin_amdgcn_wmma_f32_16x16x4_f32(
          false, a2, false, bw2[q], (short)0, c2, false, false);
    }

    // pack y[16][8] through LDS -> one fully-coalesced b128 store per lane
    if (lm < HEADS_) {
#pragma unroll
      for (int v = 0; v < 8; v++)
        yb[(v + 8 * hi) * HEADS_ + lm] = c2[v];
    }
    asm volatile("s_wait_dscnt 0" ::: "memory");
    *reinterpret_cast<float4*>(ob + mt * 128 + lane * 4) =
        *reinterpret_cast<const float4*>(yb + lane * 4);
  }
}

extern "C" void kernel_launch(void* const* d_in, const int* in_sizes, int n_in,
                              void* d_out, int out_size, void* d_ws, size_t ws_size,
                              hipStream_t stream) {
  (void)in_sizes; (void)n_in; (void)out_size; (void)ws_size;
  const float*   stacks = (const float*)d_in[0];
  const uint8_t* mask   = (const uint8_t*)d_in[1];
  const float*   gamma  = (const float*)d_in[2];
  const float*   beta   = (const float*)d_in[3];
  const float*   W1     = (const float*)d_in[4];
  const float*   b1     = (const float*)d_in[5];
  const float*   W2     = (const float*)d_in[6];
  const float*   b2     = (const float*)d_in[7];
  float* out = (float*)d_out;
  float* ws  = (float*)d_ws;

  hipMemsetAsync(ws, 0, 32 * sizeof(float), stream);       // zero sx/sx2
  stats_kernel<<<4096, 256, 0, stream>>>(stacks, mask, ws);
  finalize_kernel<<<1, 256, 0, stream>>>(mask, gamma, beta, W1, b1, W2, b2, ws);
  const int rows = B_ * N_;                                 // 16384 waves
  fused_mlp_kernel<<<rows / 8, 256, 0, stream>>>(stacks, mask, ws, out);
}